// GNN_18588618457592
// MI455X (gfx1250) — compile-verified
//
#include <hip/hip_runtime.h>
#include <hip/hip_bf16.h>
#include <math.h>

typedef float v2f __attribute__((ext_vector_type(2)));
typedef float v8f __attribute__((ext_vector_type(8)));

// ---------------------------------------------------------------------------
// Atomic helpers
// ---------------------------------------------------------------------------
__device__ __forceinline__ void atomicAddF32(float* p, float v) {
    // hardware GLOBAL_ATOMIC_ADD_F32 (no CAS loop)
    unsafeAtomicAdd(p, v);
}

__device__ __forceinline__ void atomicMaxF32(float* addr, float val) {
    // IEEE bit-punning trick: signed max for >=0, unsigned min for <0
    if (val >= 0.0f) {
        atomicMax((int*)addr, __float_as_int(val));
    } else {
        atomicMin((unsigned int*)addr, __float_as_uint(val));
    }
}

// src/dst for edge e; self-loops appended after the E input edges
__device__ __forceinline__ void edge_sd(const int* __restrict__ ei, int E, int e,
                                        int& s, int& d) {
    if (e < E) { s = ei[e]; d = ei[E + e]; }
    else       { s = e - E; d = e - E; }
}

// ---------------------------------------------------------------------------
// 0) init: agg = 0, emax = -inf, denom = 0
// ---------------------------------------------------------------------------
__global__ void gat_init_kernel(float* __restrict__ agg,
                                float* __restrict__ emax,
                                float* __restrict__ denom,
                                int n_agg, int n_nodes2) {
    int stride = gridDim.x * blockDim.x;
    for (int i = blockIdx.x * blockDim.x + threadIdx.x; i < n_agg; i += stride)
        agg[i] = 0.0f;
    for (int i = blockIdx.x * blockDim.x + threadIdx.x; i < n_nodes2; i += stride) {
        emax[i]  = -INFINITY;
        denom[i] = 0.0f;
    }
}

// ---------------------------------------------------------------------------
// 1,6) fp32 WMMA GEMM:  C[M,NDIM] = A[M,K] @ B[K,NDIM] (+ bias)
//   256 threads = 8 waves. COL_TILES = NDIM/16 waves per row-group,
//   ROWS = 16 * (8/COL_TILES) rows per block.
//   A tile staged in LDS with +1 float row padding (row stride 129 -> no
//   bank conflicts on the K-strided per-lane reads).
// ---------------------------------------------------------------------------
template <int K, int NDIM, int ROWS>
__global__ void wmma_gemm_kernel(const float* __restrict__ A,
                                 const float* __restrict__ B,
                                 const float* __restrict__ bias,
                                 float* __restrict__ C,
                                 int M) {
    constexpr int COL_TILES = NDIM / 16;
    __shared__ float Alds[ROWS * (K + 1)];

    const int tid     = threadIdx.x;
    const int wave    = tid >> 5;
    const int lane    = tid & 31;
    const int rowbase = blockIdx.x * ROWS;

    // cooperative, coalesced A-tile load (zero-fill past M)
    for (int idx = tid; idx < ROWS * K; idx += blockDim.x) {
        int r  = idx / K;
        int c  = idx - r * K;
        int gr = rowbase + r;
        Alds[r * (K + 1) + c] = (gr < M) ? A[(size_t)gr * K + c] : 0.0f;
    }
    __syncthreads();

    const int rg      = wave / COL_TILES;        // row group within block
    const int colbase = (wave % COL_TILES) * 16; // 16-col tile
    const int half    = lane >> 4;               // 0: K={0,1}, 1: K={2,3}
    const int l15     = lane & 15;

    const float* Arow = &Alds[(rg * 16 + l15) * (K + 1)];

    v8f acc = {};
    #pragma unroll 8
    for (int k = 0; k < K; k += 4) {
        const int kk = k + 2 * half;
        v2f a, b;
        a.x = Arow[kk];
        a.y = Arow[kk + 1];
        b.x = B[(size_t)kk * NDIM + colbase + l15];
        b.y = B[(size_t)(kk + 1) * NDIM + colbase + l15];
        acc = __builtin_amdgcn_wmma_f32_16x16x4_f32(
            /*neg_a=*/false, a, /*neg_b=*/false, b,
            /*c_mod=*/(short)0, acc, /*reuse_a=*/false, /*reuse_b=*/false);
    }

    // D layout: VGPR r, lanes 0-15 -> row r, lanes 16-31 -> row r+8
    const float bv = bias ? bias[colbase + l15] : 0.0f;
    #pragma unroll
    for (int r = 0; r < 8; ++r) {
        int grow = rowbase + rg * 16 + r + 8 * half;
        if (grow < M)
            C[(size_t)grow * NDIM + colbase + l15] = acc[r] + bv;
    }
}

// ---------------------------------------------------------------------------
// 2) per-node attention coefficients: alpha_{l,r}[n][h] = <xh[n,h,:], att_{l,r}[h,:]>
//    wave per node; lane covers channels {l, l+32, l+64, l+96}
// ---------------------------------------------------------------------------
__global__ void alpha_kernel(const float* __restrict__ xh,
                             const float* __restrict__ attl,
                             const float* __restrict__ attr,
                             float* __restrict__ al,
                             float* __restrict__ ar, int N) {
    const int gw   = (blockIdx.x * blockDim.x + threadIdx.x) >> 5;
    const int lane = threadIdx.x & 31;
    if (gw >= N) return;

    const float* xr = xh + (size_t)gw * 128;
    const float x0 = xr[lane], x1 = xr[lane + 32], x2 = xr[lane + 64], x3 = xr[lane + 96];

    float l0 = x0 * attl[lane]      + x1 * attl[lane + 32];
    float l1 = x2 * attl[lane + 64] + x3 * attl[lane + 96];
    float r0 = x0 * attr[lane]      + x1 * attr[lane + 32];
    float r1 = x2 * attr[lane + 64] + x3 * attr[lane + 96];

    #pragma unroll
    for (int off = 16; off > 0; off >>= 1) {
        l0 += __shfl_xor(l0, off, 32);
        l1 += __shfl_xor(l1, off, 32);
        r0 += __shfl_xor(r0, off, 32);
        r1 += __shfl_xor(r1, off, 32);
    }
    if (lane == 0) {
        al[gw * 2 + 0] = l0;  al[gw * 2 + 1] = l1;
        ar[gw * 2 + 0] = r0;  ar[gw * 2 + 1] = r1;
    }
}

// ---------------------------------------------------------------------------
// 3) edge logits + LeakyReLU + segment max over dst
// ---------------------------------------------------------------------------
__global__ void edge_logits_kernel(const int* __restrict__ ei, int E, int N,
                                   const float* __restrict__ al,
                                   const float* __restrict__ ar,
                                   float* __restrict__ elog,
                                   float* __restrict__ emax) {
    const int e = blockIdx.x * blockDim.x + threadIdx.x;
    if (e >= E + N) return;
    int s, d;
    edge_sd(ei, E, e, s, d);
    #pragma unroll
    for (int h = 0; h < 2; ++h) {
        float v = al[s * 2 + h] + ar[d * 2 + h];
        v = (v > 0.0f) ? v : 0.2f * v;          // LeakyReLU(0.2)
        elog[(size_t)e * 2 + h] = v;
        atomicMaxF32(&emax[d * 2 + h], v);
    }
}

// ---------------------------------------------------------------------------
// 4) ex = exp(e - emax[dst]) (in place) + segment sum into denom
// ---------------------------------------------------------------------------
__global__ void edge_exp_kernel(const int* __restrict__ ei, int E, int N,
                                float* __restrict__ elog,
                                const float* __restrict__ emax,
                                float* __restrict__ denom) {
    const int e = blockIdx.x * blockDim.x + threadIdx.x;
    if (e >= E + N) return;
    int s, d;
    edge_sd(ei, E, e, s, d);
    (void)s;
    #pragma unroll
    for (int h = 0; h < 2; ++h) {
        float m = emax[d * 2 + h];
        if (!(fabsf(m) <= 3.4e38f)) m = 0.0f;   // where(isfinite, m, 0)
        float v = __expf(elog[(size_t)e * 2 + h] - m);
        elog[(size_t)e * 2 + h] = v;
        atomicAddF32(&denom[d * 2 + h], v);
    }
}

// ---------------------------------------------------------------------------
// 5) weighted scatter-add: agg[dst] += xh[src] * alpha   (wave per edge)
// ---------------------------------------------------------------------------
__global__ void aggregate_kernel(const int* __restrict__ ei, int E, int N,
                                 const float* __restrict__ xh,
                                 const float* __restrict__ ex,
                                 const float* __restrict__ denom,
                                 float* __restrict__ agg) {
    const int gw   = (blockIdx.x * blockDim.x + threadIdx.x) >> 5;
    const int lane = threadIdx.x & 31;
    if (gw >= E + N) return;
    int s, d;
    edge_sd(ei, E, gw, s, d);

    const float w0 = ex[(size_t)gw * 2 + 0] / (denom[d * 2 + 0] + 1e-16f);
    const float w1 = ex[(size_t)gw * 2 + 1] / (denom[d * 2 + 1] + 1e-16f);

    const float* xs = xh  + (size_t)s * 128;
    float*       od = agg + (size_t)d * 128;
    #pragma unroll
    for (int q = 0; q < 4; ++q) {
        const int c = lane + q * 32;
        const float w = (c < 64) ? w0 : w1;     // head = c/64
        atomicAddF32(od + c, xs[c] * w);
    }
}

// ---------------------------------------------------------------------------
// launch
// ---------------------------------------------------------------------------
extern "C" void kernel_launch(void* const* d_in, const int* in_sizes, int n_in,
                              void* d_out, int out_size, void* d_ws, size_t ws_size,
                              hipStream_t stream) {
    const float* x    = (const float*)d_in[0];
    const int*   ei   = (const int*)d_in[1];
    const float* Wl   = (const float*)d_in[2];
    const float* attl = (const float*)d_in[3];
    const float* attr = (const float*)d_in[4];
    const float* Wout = (const float*)d_in[5];
    const float* bias = (const float*)d_in[6];
    float* out = (float*)d_out;

    const int N   = in_sizes[0] / 128;   // 50000 nodes
    const int E   = in_sizes[1] / 2;     // 800000 edges
    const int EDG = E + N;               // + self-loops

    // workspace layout (floats); total ~60 MB, L2-resident
    float* ws    = (float*)d_ws;
    float* xh    = ws;                         // N*128
    float* agg   = xh  + (size_t)N * 128;      // N*128
    float* al    = agg + (size_t)N * 128;      // N*2
    float* ar    = al  + (size_t)N * 2;        // N*2
    float* emax  = ar  + (size_t)N * 2;        // N*2
    float* denom = emax + (size_t)N * 2;       // N*2
    float* elog  = denom + (size_t)N * 2;      // EDG*2 (logits, then exp in place)

    gat_init_kernel<<<4096, 256, 0, stream>>>(agg, emax, denom, N * 128, N * 2);

    // xh = x @ Wl   [N,128] = [N,128]@[128,128]
    wmma_gemm_kernel<128, 128, 16><<<(N + 15) / 16, 256, 0, stream>>>(
        x, Wl, nullptr, xh, N);

    alpha_kernel<<<(N + 7) / 8, 256, 0, stream>>>(xh, attl, attr, al, ar, N);

    edge_logits_kernel<<<(EDG + 255) / 256, 256, 0, stream>>>(
        ei, E, N, al, ar, elog, emax);

    edge_exp_kernel<<<(EDG + 255) / 256, 256, 0, stream>>>(
        ei, E, N, elog, emax, denom);

    aggregate_kernel<<<(EDG + 7) / 8, 256, 0, stream>>>(
        ei, E, N, xh, elog, denom, agg);

    // out = agg @ Wout + bias   [N,64] = [N,128]@[128,64]
    wmma_gemm_kernel<128, 64, 32><<<(N + 31) / 32, 256, 0, stream>>>(
        agg, Wout, bias, out, N);
}